// MoE_66202625901112
// MI455X (gfx1250) — compile-verified
//
#include <hip/hip_runtime.h>

// MoE depthwise-conv mixture for MI455X (gfx1250, wave32).
// Roofline: ~130MB HBM traffic (~5.6us at 23.3 TB/s); VALU ~3.3 G-FMA.
// Only the routing projection is a real GEMM -> V_WMMA_F32_16X16X4_F32,
// parallelized split-K across 32 waves with f32 atomic reduction.
// Key optimization vs reference: only 2 selected experts + shared computed,
// exp/log combine fused into conv epilogue (avoids 2.1GB intermediate).

#define B_   16
#define C_   64
#define H_   128
#define W_   128
#define HW_  (H_ * W_)
#define E_   8
#define TILE 16
#define TPAD 22      // TILE + 6 halo
#define CG   16      // channels per LDS group
#define KSPLIT 32    // split-K chunks for the routing GEMM

typedef __attribute__((ext_vector_type(2))) float v2f;
typedef __attribute__((ext_vector_type(8))) float v8f;

// ---------------- Kernel 1: router conv (C=64 -> 1, 7x7, same pad) ----------
__global__ __launch_bounds__(256) void moe_router_conv(
    const float* __restrict__ x, const float* __restrict__ rw,
    const float* __restrict__ rb, float* __restrict__ r_flat)
{
    __shared__ float s_in[CG][TPAD * TPAD];
    __shared__ float s_w[CG][49];
    const int tile = blockIdx.x;                 // 64 tiles of 16x16
    const int b    = blockIdx.y;
    const int ty0  = (tile / (W_ / TILE)) * TILE;
    const int tx0  = (tile % (W_ / TILE)) * TILE;
    const int t    = threadIdx.x;                // 256 threads = one pixel each
    const int py   = t / TILE, px = t % TILE;
    float acc = 0.f;
    for (int cg = 0; cg < C_ / CG; ++cg) {
        __syncthreads();
        for (int i = t; i < CG * 49; i += 256) {
            int c = cg * CG + i / 49;
            s_w[i / 49][i % 49] = rw[c * 49 + (i % 49)];
        }
        for (int i = t; i < CG * TPAD * TPAD; i += 256) {
            int c = i / (TPAD * TPAD);
            int r = i % (TPAD * TPAD);
            int iy = ty0 - 3 + r / TPAD;
            int ix = tx0 - 3 + r % TPAD;
            float v = 0.f;
            if (iy >= 0 && iy < H_ && ix >= 0 && ix < W_)
                v = x[((size_t)(b * C_ + cg * CG + c) * H_ + iy) * W_ + ix];
            s_in[c][r] = v;
        }
        __syncthreads();
        for (int c = 0; c < CG; ++c) {
            #pragma unroll
            for (int dy = 0; dy < 7; ++dy)
                #pragma unroll
                for (int dx = 0; dx < 7; ++dx)
                    acc = fmaf(s_in[c][(py + dy) * TPAD + px + dx],
                               s_w[c][dy * 7 + dx], acc);
        }
    }
    r_flat[b * HW_ + (ty0 + py) * W_ + (tx0 + px)] = acc + rb[0];
}

// ---------------- Kernel 2a: zero the padded logits accumulator -------------
__global__ void moe_zero_logits(float* __restrict__ logits)
{
    logits[threadIdx.x] = 0.f;                  // 256 = 16x16 tile
}

// ---------------- Kernel 2b: logits = r_flat[16,16384] @ w_gate[16384,8] ----
// Split-K: KSPLIT waves each run K/KSPLIT through V_WMMA_F32_16X16X4_F32 and
// reduce their 16x16 partial tile with global_atomic_add_f32.
// A 16x4 f32 layout: lanes 0-15 M=lane (K=0/1 in v[0]/v[1]),
//                    lanes 16-31 M=lane-16 (K=2/3). B mirrors with N across lanes.
__global__ void moe_logits_wmma(const float* __restrict__ r_flat,
                                const float* __restrict__ w_gate,
                                float* __restrict__ logits)
{
    const int lane = threadIdx.x & 31;
    const int mn   = lane & 15;                 // M for A, N for B/D
    const int koff = (lane < 16) ? 0 : 2;
    const int kbeg = blockIdx.x * (HW_ / KSPLIT);
    const int kend = kbeg + (HW_ / KSPLIT);

    // Unconditional loads from a clamped column + 0/1 mask: no exec divergence
    // around the global_load in the hot loop (B cols 8..15 are zero padding).
    const float msk = (mn < E_) ? 1.f : 0.f;
    const float* __restrict__ wg = w_gate + ((mn < E_) ? mn : 0);
    const float* __restrict__ ra = r_flat + (size_t)mn * HW_;

    v8f c = {};
    for (int k0 = kbeg; k0 < kend; k0 += 4) {
        v2f a, bm;
        a.x  = ra[k0 + koff];
        a.y  = ra[k0 + koff + 1];
        bm.x = wg[(size_t)(k0 + koff) * E_] * msk;
        bm.y = wg[(size_t)(k0 + koff + 1) * E_] * msk;
        c = __builtin_amdgcn_wmma_f32_16x16x4_f32(
                false, a, false, bm, (short)0, c, false, false);
    }
    // D: VGPR r holds M = r + (lane<16 ? 0 : 8), N = lane&15
    const int mbase = (lane < 16) ? 0 : 8;
    #pragma unroll
    for (int r = 0; r < 8; ++r)
        atomicAdd(&logits[(mbase + r) * 16 + mn], c[r]);
}

// ---------------- Kernel 3: top-2 gating + CV^2 load-balancing loss ---------
__global__ void moe_gating(const float* __restrict__ logits,
                           int* __restrict__ top_idx,
                           float* __restrict__ loss_out)
{
    __shared__ float s_g[B_][2];
    __shared__ int   s_i[B_][2];
    const int t = threadIdx.x;
    if (t < B_) {
        float b1 = -1e30f, b2 = -1e30f; int i1 = 0, i2 = 0;
        for (int e = 0; e < E_; ++e) {
            float v = logits[t * 16 + e];
            if (v > b1)      { b2 = b1; i2 = i1; b1 = v; i1 = e; }
            else if (v > b2) { b2 = v; i2 = e; }
        }
        float e1 = 1.f, e2 = expf(b2 - b1);       // softmax over top-2
        float inv = 1.f / (e1 + e2);
        s_g[t][0] = e1 * inv; s_g[t][1] = e2 * inv;
        s_i[t][0] = i1;       s_i[t][1] = i2;
        top_idx[t * 2 + 0] = i1; top_idx[t * 2 + 1] = i2;
    }
    __syncthreads();
    if (t == 0) {
        float imp[E_] = {}; float load[E_] = {};
        for (int b = 0; b < B_; ++b) {
            imp[s_i[b][0]]  += s_g[b][0];
            imp[s_i[b][1]]  += s_g[b][1];
            load[s_i[b][0]] += 1.f;
            load[s_i[b][1]] += 1.f;
        }
        float mi = 0.f, ml = 0.f;
        for (int e = 0; e < E_; ++e) { mi += imp[e]; ml += load[e]; }
        mi /= E_; ml /= E_;
        float vi = 0.f, vl = 0.f;
        for (int e = 0; e < E_; ++e) {
            vi += (imp[e] - mi) * (imp[e] - mi);
            vl += (load[e] - ml) * (load[e] - ml);
        }
        vi /= (E_ - 1); vl /= (E_ - 1);           // unbiased (ddof=1)
        loss_out[0] = (vi / (mi * mi + 1e-10f) + vl / (ml * ml + 1e-10f)) * 1e-2f;
    }
}

// ---------------- Kernel 4: fused selected-expert convs + combine + shared --
__global__ __launch_bounds__(256) void moe_experts_combine(
    const float* __restrict__ x,
    const float* __restrict__ expert_w, const float* __restrict__ expert_b,
    const float* __restrict__ shared_w, const float* __restrict__ shared_b,
    const int* __restrict__ top_idx, float* __restrict__ y)
{
    __shared__ float s_in[CG][TPAD * TPAD];
    __shared__ float s_w[3][CG][49];
    __shared__ float s_bias[3][CG];
    const int tile = blockIdx.x;
    const int b    = blockIdx.y;
    const int e1   = top_idx[b * 2 + 0];
    const int e2   = top_idx[b * 2 + 1];
    const int ty0  = (tile / (W_ / TILE)) * TILE;
    const int tx0  = (tile % (W_ / TILE)) * TILE;
    const int t    = threadIdx.x;
    const int cl   = t % CG;     // channel within group
    const int ps   = t / CG;     // pixel slot (16 slots)

    __builtin_prefetch(&x[((size_t)(b * C_) * H_ + ty0) * W_ + tx0], 0, 1);

    for (int cg = 0; cg < C_ / CG; ++cg) {
        __syncthreads();
        for (int i = t; i < 3 * CG * 49; i += 256) {
            int which = i / (CG * 49);
            int rem   = i % (CG * 49);
            int c     = cg * CG + rem / 49;
            int k     = rem % 49;
            const float* wp = (which == 0) ? &expert_w[(size_t)(e1 * C_ + c) * 49]
                            : (which == 1) ? &expert_w[(size_t)(e2 * C_ + c) * 49]
                                           : &shared_w[c * 49];
            s_w[which][rem / 49][k] = wp[k];
        }
        if (t < 3 * CG) {
            int which = t / CG; int c = cg * CG + t % CG;
            s_bias[which][t % CG] = (which == 0) ? expert_b[e1 * C_ + c]
                                  : (which == 1) ? expert_b[e2 * C_ + c]
                                                 : shared_b[c];
        }
        for (int i = t; i < CG * TPAD * TPAD; i += 256) {
            int c = i / (TPAD * TPAD);
            int r = i % (TPAD * TPAD);
            int iy = ty0 - 3 + r / TPAD;
            int ix = tx0 - 3 + r % TPAD;
            float v = 0.f;
            if (iy >= 0 && iy < H_ && ix >= 0 && ix < W_)
                v = x[((size_t)(b * C_ + cg * CG + c) * H_ + iy) * W_ + ix];
            s_in[c][r] = v;
        }
        __syncthreads();
        for (int pp = 0; pp < TILE * TILE; pp += 16) {
            int pix = pp + ps;
            int py = pix / TILE, px = pix % TILE;
            float o1 = s_bias[0][cl], o2 = s_bias[1][cl], os = s_bias[2][cl];
            #pragma unroll
            for (int dy = 0; dy < 7; ++dy) {
                #pragma unroll
                for (int dx = 0; dx < 7; ++dx) {
                    float v = s_in[cl][(py + dy) * TPAD + px + dx];
                    o1 = fmaf(v, s_w[0][cl][dy * 7 + dx], o1);
                    o2 = fmaf(v, s_w[1][cl][dy * 7 + dx], o2);
                    os = fmaf(v, s_w[2][cl][dy * 7 + dx], os);
                }
            }
            float comb = expf(o1) + expf(o2);
            comb = (comb == 0.f) ? 2.220446049250313e-16f : comb;
            float val = logf(comb) + os;
            int p = (ty0 + py) * W_ + (tx0 + px);
            y[((size_t)b * HW_ + p) * C_ + cg * CG + cl] = val;
        }
    }
}

extern "C" void kernel_launch(void* const* d_in, const int* in_sizes, int n_in,
                              void* d_out, int out_size, void* d_ws, size_t ws_size,
                              hipStream_t stream) {
    const float* x        = (const float*)d_in[0];
    const float* router_w = (const float*)d_in[1];
    const float* router_b = (const float*)d_in[2];
    const float* w_gate   = (const float*)d_in[3];
    const float* expert_w = (const float*)d_in[4];
    const float* expert_b = (const float*)d_in[5];
    const float* shared_w = (const float*)d_in[6];
    const float* shared_b = (const float*)d_in[7];
    float* out = (float*)d_out;
    float* ws  = (float*)d_ws;

    float* r_flat  = ws;                       // [16, 16384]
    float* logits  = ws + (size_t)B_ * HW_;    // [16, 16] (N padded to 16)
    int*   top_idx = (int*)(logits + 256);     // [16, 2]
    float* loss    = out + (size_t)B_ * HW_ * C_;

    dim3 grid((H_ / TILE) * (W_ / TILE), B_);  // 64 tiles x 16 batches
    moe_router_conv<<<grid, 256, 0, stream>>>(x, router_w, router_b, r_flat);
    moe_zero_logits<<<1, 256, 0, stream>>>(logits);
    moe_logits_wmma<<<KSPLIT, 32, 0, stream>>>(r_flat, w_gate, logits);
    moe_gating<<<1, 32, 0, stream>>>(logits, top_idx, loss);
    moe_experts_combine<<<grid, 256, 0, stream>>>(x, expert_w, expert_b,
                                                  shared_w, shared_b, top_idx, out);
}